// VectorQuantizer_18270790877666
// MI455X (gfx1250) — compile-verified
//
#include <hip/hip_runtime.h>
#include <hip/hip_bf16.h>

typedef __attribute__((ext_vector_type(2))) float v2f;
typedef __attribute__((ext_vector_type(8))) float v8f;

#define NUM_CODES 8192
#define CODE_DIM  256
#define N_ROWS    16384
#define LDA       260   // padded LDS stride (260 % 64 == 4 -> conflict-free column reads)

// -------------------------------------------------------------------------
// 1) per-code squared norms: cnorm[k] = ||codebook[k]||^2   (one wave / row)
// -------------------------------------------------------------------------
__global__ __launch_bounds__(256) void vq_cnorm_kernel(const float* __restrict__ cb,
                                                       float* __restrict__ cnorm) {
  const int lane = threadIdx.x & 31, wave = threadIdx.x >> 5;
  const int row = blockIdx.x * 8 + wave;
  const float4* p = (const float4*)(cb + (long)row * CODE_DIM);
  float s = 0.f;
  #pragma unroll
  for (int i = lane; i < CODE_DIM / 4; i += 32) {
    float4 v = p[i];
    s += v.x * v.x + v.y * v.y + v.z * v.z + v.w * v.w;
  }
  for (int off = 16; off; off >>= 1) s += __shfl_xor(s, off, 32);
  if (lane == 0) cnorm[row] = s;
}

// -------------------------------------------------------------------------
// 2) K-group permute pack: cbp[n][4g + {0,1,2,3}] = cb[n][4g + {0,2,1,3}]
//    After packing, a WMMA B-fragment (v0=B(d+h,n), v1=B(d+2+h,n)) is one
//    contiguous b64 at cbp + n*256 + d + 2*h  -> single coalesced load.
// -------------------------------------------------------------------------
__global__ __launch_bounds__(256) void vq_pack_kernel(const float* __restrict__ cb,
                                                      float* __restrict__ cbp) {
  const long i = (long)blockIdx.x * 256 + threadIdx.x;   // float4 index
  const float4 v = ((const float4*)cb)[i];
  float4 o;
  o.x = v.x; o.y = v.z; o.z = v.y; o.w = v.w;
  ((float4*)cbp)[i] = o;
}

// -------------------------------------------------------------------------
// 3) WMMA f32 GEMM + argmin.
//    Block = 32 rows (two 16-row A tiles sharing every B fragment),
//    8 waves each own a 16-code column slice (stride 128 through K=8192).
//    Steady state: 2 wmma per (1 global b64 + 1 ds_load_2addr_b64).
//    PACKED:   B fragment = one b64 from group-permuted codebook (coalesced)
//    !PACKED:  two strided b32 loads from the raw codebook (ws fallback)
// -------------------------------------------------------------------------
template <bool PACKED>
__global__ __launch_bounds__(256) void vq_argmin_kernel(const float* __restrict__ z,
                                                        const float* __restrict__ Bm,
                                                        const float* __restrict__ cnorm,
                                                        int* __restrict__ out_idx) {
  __shared__ float As[32 * LDA];
  __shared__ float rv[8][32];
  __shared__ int   ri[8][32];

  const int tid = threadIdx.x, lane = tid & 31, wave = tid >> 5;
  const int rowBase = blockIdx.x << 5;    // 32 rows per block

  // stage A tile: 32 rows x 256 dims, float4-coalesced into padded LDS
  {
    const float4* zp = (const float4*)(z + (long)rowBase * CODE_DIM);
    #pragma unroll
    for (int i = tid; i < 32 * (CODE_DIM / 4); i += 256) {
      const int r = i >> 6, cq = i & 63;
      float4 v = zp[(r << 6) + cq];
      *(float4*)&As[r * LDA + (cq << 2)] = v;
    }
  }
  __syncthreads();

  const int nlo = lane & 15;     // N (codes) / M (rows) sub-lane
  const int half = lane >> 4;    // which 16-lane half
  const float* ap0 = &As[nlo * LDA + (half << 1)];            // rows 0..15
  const float* ap1 = ap0 + 16 * LDA;                          // rows 16..31

  float minv0[8], minv1[8];
  int   mini0[8], mini1[8];
  #pragma unroll
  for (int r = 0; r < 8; ++r) {
    minv0[r] = __builtin_inff(); mini0[r] = 0;
    minv1[r] = __builtin_inff(); mini1[r] = 0;
  }

  for (int nb = wave * 16; nb < NUM_CODES; nb += 128) {
    const int n = nb + nlo;
    const float cn = cnorm[n];
    const float* bp = Bm + (long)n * CODE_DIM;    // this lane's code row
    if (nb + 128 < NUM_CODES)
      __builtin_prefetch(bp + (long)128 * CODE_DIM, 0, 3);  // next chunk hint

    v8f c0 = {}, c1 = {};
    #pragma unroll 8
    for (int d = 0; d < CODE_DIM; d += 4) {
      v2f b;
      if constexpr (PACKED) {
        // packed group: {B(d+h,n), B(d+2+h,n)} contiguous -> one b64
        b = *(const v2f*)(bp + d + (half << 1));
      } else {
        b.x = bp[d + half];
        b.y = bp[d + half + 2];
      }
      // A frags: lanes 0-15 -> {A[m][d],A[m][d+1]}, lanes 16-31 -> {A[m][d+2],A[m][d+3]}
      v2f a0 = *(const v2f*)(ap0 + d);
      v2f a1 = *(const v2f*)(ap1 + d);
      c0 = __builtin_amdgcn_wmma_f32_16x16x4_f32(false, a0, false, b, (short)0, c0,
                                                 false, false);
      c1 = __builtin_amdgcn_wmma_f32_16x16x4_f32(false, a1, false, b, (short)0, c1,
                                                 false, false);
    }

    // argmin key: ||e||^2 - 2 * (z . e)   (per-lane n strictly increases -> '<' keeps first)
    #pragma unroll
    for (int r = 0; r < 8; ++r) {
      const float v0 = fmaf(-2.f, c0[r], cn);
      if (v0 < minv0[r]) { minv0[r] = v0; mini0[r] = n; }
      const float v1 = fmaf(-2.f, c1[r], cn);
      if (v1 < minv1[r]) { minv1[r] = v1; mini1[r] = n; }
    }
  }

  // reduce across the 16 lanes of each half (butterfly, first-index tie-break)
  #pragma unroll
  for (int r = 0; r < 8; ++r) {
    float mv0 = minv0[r]; int mi0 = mini0[r];
    float mv1 = minv1[r]; int mi1 = mini1[r];
    for (int off = 8; off; off >>= 1) {
      const float ov0 = __shfl_xor(mv0, off, 32);
      const int   oi0 = __shfl_xor(mi0, off, 32);
      if (ov0 < mv0 || (ov0 == mv0 && oi0 < mi0)) { mv0 = ov0; mi0 = oi0; }
      const float ov1 = __shfl_xor(mv1, off, 32);
      const int   oi1 = __shfl_xor(mi1, off, 32);
      if (ov1 < mv1 || (ov1 == mv1 && oi1 < mi1)) { mv1 = ov1; mi1 = oi1; }
    }
    if (nlo == 0) {  // lane 0 -> rows r/r+16, lane 16 -> rows r+8/r+24
      rv[wave][r + (half << 3)]      = mv0;
      ri[wave][r + (half << 3)]      = mi0;
      rv[wave][r + (half << 3) + 16] = mv1;
      ri[wave][r + (half << 3) + 16] = mi1;
    }
  }
  __syncthreads();

  // combine the 8 waves' candidates per row (fixed order -> deterministic)
  if (tid < 32) {
    float bv = rv[0][tid]; int bi = ri[0][tid];
    #pragma unroll
    for (int w = 1; w < 8; ++w) {
      const float v = rv[w][tid]; const int iw = ri[w][tid];
      if (v < bv || (v == bv && iw < bi)) { bv = v; bi = iw; }
    }
    out_idx[rowBase + tid] = bi;
  }
}

// -------------------------------------------------------------------------
// 4) gather z_q = codebook[idx], per-row squared-error partial sums
// -------------------------------------------------------------------------
__global__ __launch_bounds__(256) void vq_gather_kernel(const float* __restrict__ z,
                                                        const float* __restrict__ cb,
                                                        const int* __restrict__ idx,
                                                        float* __restrict__ zq,
                                                        float* __restrict__ partial) {
  const int lane = threadIdx.x & 31, wave = threadIdx.x >> 5;
  const int row = blockIdx.x * 8 + wave;
  const int ci = idx[row];
  const float4* ep = (const float4*)(cb + (long)ci * CODE_DIM);
  const float4* zp = (const float4*)(z + (long)row * CODE_DIM);
  float4*       op = (float4*)(zq + (long)row * CODE_DIM);
  float s = 0.f;
  #pragma unroll
  for (int i = lane; i < CODE_DIM / 4; i += 32) {
    const float4 ev = ep[i], zv = zp[i];
    op[i] = ev;
    const float dx = zv.x - ev.x, dy = zv.y - ev.y, dz = zv.z - ev.z, dw = zv.w - ev.w;
    s += dx * dx + dy * dy + dz * dz + dw * dw;
  }
  for (int off = 16; off; off >>= 1) s += __shfl_xor(s, off, 32);
  if (lane == 0) partial[row] = s;
}

// -------------------------------------------------------------------------
// 5) deterministic final reduction: vq_loss = (1 + BETA) * MSE
// -------------------------------------------------------------------------
__global__ __launch_bounds__(256) void vq_loss_kernel(const float* __restrict__ partial,
                                                      float* __restrict__ out_loss) {
  __shared__ float sm[256];
  float s = 0.f;
  for (int i = threadIdx.x; i < N_ROWS; i += 256) s += partial[i];
  sm[threadIdx.x] = s;
  __syncthreads();
  for (int off = 128; off; off >>= 1) {
    if (threadIdx.x < off) sm[threadIdx.x] += sm[threadIdx.x + off];
    __syncthreads();
  }
  if (threadIdx.x == 0)
    out_loss[0] = sm[0] * (1.1f / (float)(N_ROWS * CODE_DIM));
}

// -------------------------------------------------------------------------
extern "C" void kernel_launch(void* const* d_in, const int* in_sizes, int n_in,
                              void* d_out, int out_size, void* d_ws, size_t ws_size,
                              hipStream_t stream) {
  (void)in_sizes; (void)n_in;
  const float* z  = (const float*)d_in[0];   // [16,1024,256] f32
  const float* cb = (const float*)d_in[1];   // [8192,256]   f32
  float* zq   = (float*)d_out;               // 16*1024*256 floats
  float* loss = zq + (out_size - 1);         // final element

  char* ws = (char*)d_ws;
  float* cnorm   = (float*)ws;                  //  32 KB (8192 f32)
  int*   idx     = (int*)(ws + 32 * 1024);      //  64 KB (16384 i32)
  float* partial = (float*)(ws + 96 * 1024);    //  64 KB (16384 f32)
  float* cbp     = (float*)(ws + 160 * 1024);   //   8 MB (8192*256 f32, group-permuted)

  const bool usePacked = ws_size >= (160 * 1024 + (size_t)CODE_DIM * NUM_CODES * 4);

  vq_cnorm_kernel<<<NUM_CODES / 8, 256, 0, stream>>>(cb, cnorm);

  if (usePacked) {
    vq_pack_kernel<<<(NUM_CODES * CODE_DIM / 4) / 256, 256, 0, stream>>>(cb, cbp);
    vq_argmin_kernel<true><<<N_ROWS / 32, 256, 0, stream>>>(z, cbp, cnorm, idx);
  } else {
    vq_argmin_kernel<false><<<N_ROWS / 32, 256, 0, stream>>>(z, cb, cnorm, idx);
  }

  vq_gather_kernel<<<N_ROWS / 8, 256, 0, stream>>>(z, cb, idx, zq, partial);
  vq_loss_kernel<<<1, 256, 0, stream>>>(partial, loss);
}